// GraphContrastiveWithNegatives_87411174408699
// MI455X (gfx1250) — compile-verified
//
#include <hip/hip_runtime.h>
#include <stdint.h>

// Problem constants (from reference)
#define NN   2048      // nodes
#define DD   512       // embedding dim
#define EE   32768     // edges
#define KK   5         // negatives per edge
#define TEMP_INV 10.0f // 1 / 0.1

typedef __attribute__((ext_vector_type(2))) float v2f;
typedef __attribute__((ext_vector_type(8))) float v8f;

#define NEG_INF (-__builtin_inff())

// ---------------------------------------------------------------------------
// Kernel 1: L2-normalize rows (divide by max(||x||, 1e-12), like F.normalize)
// One block (256 threads) per row of 512 floats.
// ---------------------------------------------------------------------------
__global__ void k_normalize_rows(const float* __restrict__ X, float* __restrict__ Y) {
  const int row = blockIdx.x;
  const int t = threadIdx.x;              // 256 threads
  const float* xr = X + (size_t)row * DD;
  float a = xr[t];
  float b = xr[t + 256];
  __shared__ float red[256];
  red[t] = a * a + b * b;
  __syncthreads();
  for (int off = 128; off > 0; off >>= 1) {
    if (t < off) red[t] += red[t + off];
    __syncthreads();
  }
  const float nrm = sqrtf(red[0]);
  const float inv = 1.0f / fmaxf(nrm, 1e-12f);
  Y[(size_t)row * DD + t]       = a * inv;
  Y[(size_t)row * DD + t + 256] = b * inv;
}

// ---------------------------------------------------------------------------
// Kernel 2: zero adjacency bitmask (N*N bits = 131072 u32 words) and d_out.
// ---------------------------------------------------------------------------
__global__ void k_zero(uint32_t* __restrict__ adj, float* __restrict__ out) {
  const int i = blockIdx.x * blockDim.x + threadIdx.x;
  adj[i] = 0u;
  if (i == 0) out[0] = 0.0f;
}

// ---------------------------------------------------------------------------
// Kernel 3: scatter symmetric adjacency bits.
// ---------------------------------------------------------------------------
__global__ void k_build_adj(const int* __restrict__ ei, uint32_t* __restrict__ adj) {
  const int e = blockIdx.x * blockDim.x + threadIdx.x;
  if (e >= EE) return;
  const unsigned s = (unsigned)ei[e];
  const unsigned d = (unsigned)ei[EE + e];
  const unsigned b1 = s * NN + d;
  const unsigned b2 = d * NN + s;
  atomicOr(&adj[b1 >> 5], 1u << (b1 & 31u));
  atomicOr(&adj[b2 >> 5], 1u << (b2 & 31u));
}

// ---------------------------------------------------------------------------
// Kernel 4: Gram matrix G = Xn * Xn^T via fp32 WMMA (V_WMMA_F32_16X16X4_F32).
// One wave32 per 16x16 tile of G; K loop over D in steps of 4.
// A (16x4, MxK) layout: lane = row M (both halves), VGPR0 holds K={0 | 2},
// VGPR1 holds K={1 | 3} -> each lane loads a contiguous float2.
// B (4x16, KxN) is X^T, so B[k][n] = X[n][k]: same load pattern with n0.
// 16 MB result stays resident in the 192 MB L2 for the gather phase.
// ---------------------------------------------------------------------------
__global__ void k_gram_wmma(const float* __restrict__ X, float* __restrict__ G) {
  const int n0 = blockIdx.x * 16;
  const int m0 = blockIdx.y * 16;
  const int lane = threadIdx.x;           // 32 threads = 1 wave
  const int r  = lane & 15;
  const int hi = lane >> 4;
  const float* arow = X + (size_t)(m0 + r) * DD;
  const float* brow = X + (size_t)(n0 + r) * DD;
  v8f c = {};
#pragma unroll 8
  for (int k = 0; k < DD; k += 4) {
    v2f a = *(const v2f*)(arow + k + 2 * hi);
    v2f b = *(const v2f*)(brow + k + 2 * hi);
    c = __builtin_amdgcn_wmma_f32_16x16x4_f32(
        /*neg_a=*/false, a, /*neg_b=*/false, b,
        /*c_mod=*/(short)0, c, /*reuse_a=*/false, /*reuse_b=*/false);
  }
  // D layout: VGPR v -> M = v + 8*hi, N = r
#pragma unroll
  for (int v = 0; v < 8; ++v) {
    G[(size_t)(m0 + v + 8 * hi) * NN + n0 + r] = c[v];
  }
}

// ---------------------------------------------------------------------------
// Threefry-2x32 (20 rounds), JAX schedule.
// ---------------------------------------------------------------------------
__device__ __forceinline__ uint32_t rotl32(uint32_t x, uint32_t r) {
  return (x << r) | (x >> (32u - r));
}

__device__ __forceinline__ void tf2x32(uint32_t k0, uint32_t k1,
                                       uint32_t x0, uint32_t x1,
                                       uint32_t& o0, uint32_t& o1) {
  const uint32_t k2 = k0 ^ k1 ^ 0x1BD11BDAu;
  x0 += k0; x1 += k1;
  // group 1: rotations [13,15,26,6], inject (k1, k2+1)
  x0 += x1; x1 = rotl32(x1, 13); x1 ^= x0;
  x0 += x1; x1 = rotl32(x1, 15); x1 ^= x0;
  x0 += x1; x1 = rotl32(x1, 26); x1 ^= x0;
  x0 += x1; x1 = rotl32(x1, 6);  x1 ^= x0;
  x0 += k1; x1 += k2 + 1u;
  // group 2: rotations [17,29,16,24], inject (k2, k0+2)
  x0 += x1; x1 = rotl32(x1, 17); x1 ^= x0;
  x0 += x1; x1 = rotl32(x1, 29); x1 ^= x0;
  x0 += x1; x1 = rotl32(x1, 16); x1 ^= x0;
  x0 += x1; x1 = rotl32(x1, 24); x1 ^= x0;
  x0 += k2; x1 += k0 + 2u;
  // group 3: rotations [13,15,26,6], inject (k0, k1+3)
  x0 += x1; x1 = rotl32(x1, 13); x1 ^= x0;
  x0 += x1; x1 = rotl32(x1, 15); x1 ^= x0;
  x0 += x1; x1 = rotl32(x1, 26); x1 ^= x0;
  x0 += x1; x1 = rotl32(x1, 6);  x1 ^= x0;
  x0 += k0; x1 += k1 + 3u;
  // group 4: rotations [17,29,16,24], inject (k1, k2+4)
  x0 += x1; x1 = rotl32(x1, 17); x1 ^= x0;
  x0 += x1; x1 = rotl32(x1, 29); x1 ^= x0;
  x0 += x1; x1 = rotl32(x1, 16); x1 ^= x0;
  x0 += x1; x1 = rotl32(x1, 24); x1 ^= x0;
  x0 += k1; x1 += k2 + 4u;
  // group 5: rotations [13,15,26,6], inject (k2, k0+5)
  x0 += x1; x1 = rotl32(x1, 13); x1 ^= x0;
  x0 += x1; x1 = rotl32(x1, 15); x1 ^= x0;
  x0 += x1; x1 = rotl32(x1, 26); x1 ^= x0;
  x0 += x1; x1 = rotl32(x1, 6);  x1 ^= x0;
  x0 += k2; x1 += k0 + 5u;
  o0 = x0; o1 = x1;
}

// ---------------------------------------------------------------------------
// Kernel 5: per-edge scoring, top-K negative selection, InfoNCE loss.
// One block (256 threads) per edge; each thread owns 8 of the N=2048 scores.
// All G gathers hit L2 (G is 16 MB << 192 MB).
// ---------------------------------------------------------------------------
__global__ void k_loss(const int* __restrict__ ei,
                       const float* __restrict__ G,
                       const uint32_t* __restrict__ adj,
                       float* __restrict__ out) {
  const int e = blockIdx.x;
  const int t = threadIdx.x;              // 256 threads
  __shared__ int   s_src, s_dst;
  __shared__ float sval[256];
  __shared__ int   sidx[256];
  __shared__ int   winners[KK];
  if (t == 0) { s_src = ei[e]; s_dst = ei[EE + e]; }
  __syncthreads();
  const int src = s_src;

  // JAX key: fold_in(key(0), 123) == threefry2x32([0,0], [0,123])
  uint32_t fk0, fk1;
  tf2x32(0u, 0u, 0u, 123u, fk0, fk1);

  const uint32_t half = (uint32_t)EE * (uint32_t)NN / 2u;  // 33554432
  float sc[8];
#pragma unroll
  for (int j = 0; j < 8; ++j) {
    const int node = j * 256 + t;
    const uint32_t i  = (uint32_t)e * (uint32_t)NN + (uint32_t)node;
    const uint32_t cj = (i < half) ? i : (i - half);
    uint32_t o0, o1;
    tf2x32(fk0, fk1, cj, cj + half, o0, o1);
    const uint32_t bits = (i < half) ? o0 : o1;
    const float u = __uint_as_float((bits >> 9) | 0x3F800000u) - 1.0f;
    const uint32_t b = (uint32_t)src * NN + (uint32_t)node;
    const bool nb = (adj[b >> 5] >> (b & 31u)) & 1u;
    sc[j] = nb ? NEG_INF : u;
  }

  // K rounds of block-wide argmax (ties -> lower index, matching lax.top_k)
  for (int sel = 0; sel < KK; ++sel) {
    float bv = NEG_INF;
    int   bi = NN;
#pragma unroll
    for (int j = 0; j < 8; ++j) {
      const int node = j * 256 + t;
      if (sc[j] > bv || (sc[j] == bv && node < bi)) { bv = sc[j]; bi = node; }
    }
    sval[t] = bv; sidx[t] = bi;
    __syncthreads();
    for (int off = 128; off > 0; off >>= 1) {
      if (t < off) {
        const float ov = sval[t + off];
        const int   oi = sidx[t + off];
        if (ov > sval[t] || (ov == sval[t] && oi < sidx[t])) { sval[t] = ov; sidx[t] = oi; }
      }
      __syncthreads();
    }
    const int win = sidx[0];
    if (t == 0) winners[sel] = win;
    if ((win & 255) == t) sc[win >> 8] = NEG_INF;  // remove winner
    __syncthreads();
  }

  if (t == 0) {
    const float* grow = G + (size_t)src * NN;
    float lg[KK + 1];
    lg[0] = grow[s_dst] * TEMP_INV;
#pragma unroll
    for (int k = 0; k < KK; ++k) lg[k + 1] = grow[winners[k]] * TEMP_INV;
    float m = lg[0];
#pragma unroll
    for (int k = 1; k <= KK; ++k) m = fmaxf(m, lg[k]);
    float s = 0.0f;
#pragma unroll
    for (int k = 0; k <= KK; ++k) s += expf(lg[k] - m);
    const float loss = (logf(s) + m) - lg[0];
    atomicAdd(out, loss * (1.0f / (float)EE));
  }
}

// ---------------------------------------------------------------------------
// Launch
// ---------------------------------------------------------------------------
extern "C" void kernel_launch(void* const* d_in, const int* in_sizes, int n_in,
                              void* d_out, int out_size, void* d_ws, size_t ws_size,
                              hipStream_t stream) {
  (void)in_sizes; (void)n_in; (void)out_size; (void)ws_size;
  const float* emb = (const float*)d_in[0];   // [N, D] fp32
  const int*   ei  = (const int*)d_in[1];     // [2, E] int32 (row0 = src, row1 = dst)
  float* out = (float*)d_out;                 // scalar loss

  char* ws = (char*)d_ws;
  float*    emb_n = (float*)ws;                                             // 4 MB
  float*    G     = (float*)(ws + (size_t)NN * DD * sizeof(float));         // 16 MB
  uint32_t* adj   = (uint32_t*)(ws + (size_t)NN * DD * sizeof(float)
                                   + (size_t)NN * NN * sizeof(float));      // 512 KB

  k_normalize_rows<<<NN, 256, 0, stream>>>(emb, emb_n);
  k_zero<<<(NN * NN / 32) / 256, 256, 0, stream>>>(adj, out);
  k_build_adj<<<EE / 256, 256, 0, stream>>>(ei, adj);
  k_gram_wmma<<<dim3(NN / 16, NN / 16), 32, 0, stream>>>(emb_n, G);
  k_loss<<<EE, 256, 0, stream>>>(ei, G, adj, out);
}